// SSIMLoss_67774583931370
// MI455X (gfx1250) — compile-verified
//
#include <hip/hip_runtime.h>

typedef __attribute__((ext_vector_type(2))) float v2f;
typedef __attribute__((ext_vector_type(8))) float v8f;

#define IMG   384
#define OUTD  378
#define NIMG  64
#define TILES 24      // ceil(378/16)
#define REG   22      // 16 + 7 - 1 input rows/cols per output tile
#define PADC  24      // LDS cols padded to multiple of 4 (K granularity)
#define KCH   6       // ceil(22/4) K-chunks

__global__ void ssim_zero_kernel(float* acc) { acc[0] = 0.0f; }

__global__ void ssim_final_kernel(const float* __restrict__ acc, float* __restrict__ out) {
    out[0] = -acc[0] / (float)((long)NIMG * OUTD * OUTD);
}

__global__ __launch_bounds__(32) void ssim_tile_kernel(
    const float* __restrict__ X, const float* __restrict__ Y,
    const float* __restrict__ mxv, float* __restrict__ acc_out)
{
    __shared__ float Xs[REG * PADC];
    __shared__ float Ys[REG * PADC];
    __shared__ float Hb[5][PADC][16];   // horizontal box sums, rows 22..23 stay zero

    const int lane = threadIdx.x;                 // 0..31 (wave32)
    const int tr   = blockIdx.x / TILES;
    const int tc   = blockIdx.x % TILES;
    const int n    = blockIdx.y;
    const int r0   = tr * 16, c0 = tc * 16;
    const float q  = 100000.0f;

    const float* Xn = X + (size_t)n * IMG * IMG;
    const float* Yn = Y + (size_t)n * IMG * IMG;

    // ---- stage the 22x22 input region into LDS (scaled by q, zero-padded) ----
    for (int i = lane; i < REG * PADC; i += 32) {
        int r = i / PADC, c = i % PADC;
        int gr = r0 + r, gc = c0 + c;
        float xv = 0.0f, yv = 0.0f;
        if (c < REG && gr < IMG && gc < IMG) {
            xv = Xn[gr * IMG + gc] * q;
            yv = Yn[gr * IMG + gc] * q;
        }
        Xs[i] = xv; Ys[i] = yv;
    }
    for (int i = lane; i < 5 * PADC * 16; i += 32) ((float*)Hb)[i] = 0.0f;
    __syncthreads();

    const int  bn   = lane & 15;          // N index (B/C/D) and M index (A), per half-wave
    const int  hiK  = (lane >> 4) * 2;    // K offset for upper half-wave in A/B fragments
    const bool hiM  = (lane >> 4) != 0;   // upper half-wave holds M+8 rows of C/D

    // =========== pass 1: horizontal 7-tap box sums via WMMA band matmul ===========
    // H(22x16) = In(22x24) x B(24x16),  B[k][c] = (c <= k <= c+6)
    #pragma unroll
    for (int rb = 0; rb < 2; ++rb) {
        v8f hx = {}, hy = {}, hxx = {}, hyy = {}, hxy = {};
        const int row = rb * 16 + bn;          // A-matrix row for this lane
        const bool rok = row < REG;
        #pragma unroll
        for (int kc = 0; kc < KCH; ++kc) {
            v2f ax, ay, axx, ayy, axy, bb;
            #pragma unroll
            for (int v = 0; v < 2; ++v) {
                const int k = kc * 4 + hiK + v;          // K index (col of In / row of B)
                float xv = 0.0f, yv = 0.0f;
                if (rok) { xv = Xs[row * PADC + k]; yv = Ys[row * PADC + k]; }
                ax[v]  = xv;      ay[v]  = yv;
                axx[v] = xv * xv; ayy[v] = yv * yv; axy[v] = xv * yv;
                bb[v]  = (k >= bn && k <= bn + 6) ? 1.0f : 0.0f;
            }
            hx  = __builtin_amdgcn_wmma_f32_16x16x4_f32(false, ax,  false, bb, (short)0, hx,  false, false);
            hy  = __builtin_amdgcn_wmma_f32_16x16x4_f32(false, ay,  false, bb, (short)0, hy,  false, false);
            hxx = __builtin_amdgcn_wmma_f32_16x16x4_f32(false, axx, false, bb, (short)0, hxx, false, false);
            hyy = __builtin_amdgcn_wmma_f32_16x16x4_f32(false, ayy, false, bb, (short)0, hyy, false, false);
            hxy = __builtin_amdgcn_wmma_f32_16x16x4_f32(false, axy, false, bb, (short)0, hxy, false, false);
        }
        // D layout: VGPR p -> M = p (lanes 0-15) / p+8 (lanes 16-31), N = lane&15
        const int mbase = rb * 16 + (hiM ? 8 : 0);
        #pragma unroll
        for (int p = 0; p < 8; ++p) {
            const int m = mbase + p;
            if (m < REG) {
                Hb[0][m][bn] = hx[p];
                Hb[1][m][bn] = hy[p];
                Hb[2][m][bn] = hxx[p];
                Hb[3][m][bn] = hyy[p];
                Hb[4][m][bn] = hxy[p];
            }
        }
    }
    __syncthreads();

    // =========== pass 2: vertical 7-tap box sums, weight 1/49 folded in ===========
    // U(16x16) = C(16x24) x H(24x16),  C[i][k] = (i <= k <= i+6) / 49
    v8f u[5];
    #pragma unroll
    for (int m = 0; m < 5; ++m) u[m] = (v8f){};
    #pragma unroll
    for (int kc = 0; kc < KCH; ++kc) {
        v2f ca;
        v2f b0, b1, b2, b3, b4;
        #pragma unroll
        for (int v = 0; v < 2; ++v) {
            const int k = kc * 4 + hiK + v;
            ca[v] = (k >= bn && k <= bn + 6) ? (1.0f / 49.0f) : 0.0f;
            b0[v] = Hb[0][k][bn];
            b1[v] = Hb[1][k][bn];
            b2[v] = Hb[2][k][bn];
            b3[v] = Hb[3][k][bn];
            b4[v] = Hb[4][k][bn];
        }
        u[0] = __builtin_amdgcn_wmma_f32_16x16x4_f32(false, ca, false, b0, (short)0, u[0], false, false);
        u[1] = __builtin_amdgcn_wmma_f32_16x16x4_f32(false, ca, false, b1, (short)0, u[1], false, false);
        u[2] = __builtin_amdgcn_wmma_f32_16x16x4_f32(false, ca, false, b2, (short)0, u[2], false, false);
        u[3] = __builtin_amdgcn_wmma_f32_16x16x4_f32(false, ca, false, b3, (short)0, u[3], false, false);
        u[4] = __builtin_amdgcn_wmma_f32_16x16x4_f32(false, ca, false, b4, (short)0, u[4], false, false);
    }

    // =========== per-pixel SSIM on the D-fragment layout ===========
    const float mxn = mxv[n] * q;
    const float C1  = (0.01f * mxn) * (0.01f * mxn);
    const float C2  = (0.03f * mxn) * (0.03f * mxn);
    const float COV = 49.0f / 48.0f;

    float ssum = 0.0f;
    const int mbase = hiM ? 8 : 0;
    #pragma unroll
    for (int p = 0; p < 8; ++p) {
        const int orr = r0 + mbase + p;
        const int occ = c0 + bn;
        if (orr < OUTD && occ < OUTD) {
            const float ux = u[0][p], uy = u[1][p];
            const float uxx = u[2][p], uyy = u[3][p], uxy = u[4][p];
            const float vx  = COV * (uxx - ux * ux);
            const float vy  = COV * (uyy - uy * uy);
            const float vxy = COV * (uxy - ux * uy);
            const float A1 = 2.0f * ux * uy + C1;
            const float A2 = 2.0f * vxy + C2;
            const float B1 = ux * ux + uy * uy + C1;
            const float B2 = vx + vy + C2;
            ssum += (A1 * A2) / (B1 * B2);
        }
    }

    // wave32 reduction + one atomic per wave
    #pragma unroll
    for (int off = 16; off > 0; off >>= 1)
        ssum += __shfl_down(ssum, off, 32);
    if (lane == 0) atomicAdd(acc_out, ssum);
}

extern "C" void kernel_launch(void* const* d_in, const int* in_sizes, int n_in,
                              void* d_out, int out_size, void* d_ws, size_t ws_size,
                              hipStream_t stream) {
    const float* X  = (const float*)d_in[0];
    const float* Y  = (const float*)d_in[1];
    // d_in[2] = norm (unused by the reference), d_in[4] = w (constant 1/49, folded in)
    const float* mx = (const float*)d_in[3];
    float* acc = (float*)d_ws;
    float* out = (float*)d_out;

    ssim_zero_kernel<<<1, 1, 0, stream>>>(acc);
    dim3 grid(TILES * TILES, NIMG);
    ssim_tile_kernel<<<grid, 32, 0, stream>>>(X, Y, mx, acc);
    ssim_final_kernel<<<1, 1, 0, stream>>>(acc, out);
}